// FNO1d_x_to_2d_26336739459433
// MI455X (gfx1250) — compile-verified
//
#include <hip/hip_runtime.h>
#include <math.h>

typedef float v2f __attribute__((ext_vector_type(2)));
typedef float v8f __attribute__((ext_vector_type(8)));

#define NP 268          // padded spatial size (256 + 12)
#define NKY 20
#define NKX 40
#define WID 64
#define BB 8
#define TIT_LD 288      // Ti_t table padded width (>= 2*144)

__device__ __forceinline__ int lane_id() { return (int)(threadIdx.x & 31); }

// ---- WMMA f32 16x16x4 fragment helpers (per CDNA5 ISA VGPR layouts) ----
// A (16x4): lanes 0-15 hold rows M=lane at K={k,k+1}; lanes 16-31 at K={k+2,k+3}
__device__ __forceinline__ v2f frag_a(const float* __restrict__ A, int row0, int k, int ld) {
  int l = lane_id();
  const float* p = A + (size_t)(row0 + (l & 15)) * ld + k + ((l >> 4) << 1);
  return v2f{p[0], p[1]};
}
// B (4x16): VGPR0 = rows {k, k+2} striped across lane halves, VGPR1 = rows {k+1, k+3}
__device__ __forceinline__ v2f frag_b(const float* __restrict__ B, int k, int col0, int ld) {
  int l = lane_id();
  const float* p = B + (size_t)(k + ((l >> 4) << 1)) * ld + col0 + (l & 15);
  return v2f{p[0], p[ld]};
}
__device__ __forceinline__ v8f wmma4(v2f a, v2f b, v8f c) {
  return __builtin_amdgcn_wmma_f32_16x16x4_f32(false, a, false, b, (short)0, c, false, false);
}
__device__ __forceinline__ float gelu_exact(float v) {
  return 0.5f * v * (1.0f + erff(v * 0.70710678118654752f));
}

// ---- twiddle tables (built on device; deterministic, graph-capturable) ----
__global__ void twiddle_init(float* __restrict__ tf_t, float* __restrict__ tf_x,
                             float* __restrict__ ti_x, float* __restrict__ ti_t) {
  int i = blockIdx.x * blockDim.x + threadIdx.x;
  const float TWO_PI = 6.2831853071795864769f;
  if (i < NP * NKY) {              // forward t-DFT: e^{-i 2pi ky t / NP}
    int t = i / NKY, ky = i % NKY;
    float th = TWO_PI * (float)ky * (float)t / (float)NP;
    tf_t[2 * i] = __cosf(th) * 0.0f + cosf(th);
    tf_t[2 * i + 1] = -sinf(th);
  }
  if (i < NP * NKX) {              // x-DFT fwd/inv for kx in {0..19} U {248..267}
    int x = i / NKX, kx = i % NKX;
    int kv = (kx < 20) ? kx : kx + 228;
    float th = TWO_PI * (float)kv * (float)x / (float)NP;
    float c = cosf(th), s = sinf(th);
    tf_x[2 * i] = c;  tf_x[2 * i + 1] = -s;
    ti_x[2 * i] = c / (float)NP;  ti_x[2 * i + 1] = s / (float)NP;
  }
  if (i < NKX * TIT_LD) {          // t-inverse as real 40xNP matrix (rfft reconstruction)
    int j = i / TIT_LD, t = i % TIT_LD;
    float v = 0.f;
    if (t < NP) {
      float th = TWO_PI * (float)t / (float)NP;
      if (j == 0) v = 1.f / (float)NP;                               // re, ky=0
      else if (j < 20) v = 2.f * cosf(th * (float)j) / (float)NP;    // re, ky>=1
      else { int jj = j - 20; v = (jj == 0) ? 0.f : -2.f * sinf(th * (float)jj) / (float)NP; } // im
    }
    ti_t[i] = v;
  }
}

// ---- lift: h0[b,c,x,t] = w0_w[c]*(x^T wi_w^T + wi_b) + w0_b[c], zero in pad ----
__global__ void lift_kernel(const float* __restrict__ xin, const float* __restrict__ wiw,
                            const float* __restrict__ wib, const float* __restrict__ w0w,
                            const float* __restrict__ w0b, float* __restrict__ h) {
  int bid = blockIdx.x;                 // b*NP + x
  int b = bid / NP, x = bid % NP;
  for (int t = threadIdx.x; t < NP; t += 128) {
    bool in = (x < 256) && (t < 256);
    float s = 0.f;
    if (in) {
      s = wib[x];
#pragma unroll
      for (int i = 0; i < 8; ++i) s += xin[((size_t)b * 8 + i) * 256 + t] * wiw[x * 8 + i];
    }
    for (int c = 0; c < WID; ++c) {
      float v = in ? (w0w[c] * s + w0b[c]) : 0.f;
      h[(((size_t)(b * WID + c)) * NP + x) * NP + t] = v;
    }
  }
}

// ---- forward t-DFT: one block per (b,c,x) row; 268 reals -> 20 complex ----
__global__ void dft_t_fwd(const float* __restrict__ h, const float* __restrict__ tw,
                          float* __restrict__ s1) {
  size_t row = blockIdx.x;
  __shared__ float r[NP];
  const float* hp = h + row * NP;
  for (int t = threadIdx.x; t < NP; t += 64) r[t] = hp[t];
  __syncthreads();
  int tid = threadIdx.x;
  if (tid < 2 * NKY) {
    int j = tid % NKY, im = tid / NKY;
    float acc = 0.f;
    for (int t = 0; t < NP; ++t) acc += r[t] * tw[(t * NKY + j) * 2 + im];
    s1[row * (2 * NKY) + j * 2 + im] = acc;
  }
}

// ---- forward x-DFT: block per (b,c); (268 x, 20 ky) complex -> (40 kx, 20 ky) ----
__global__ void dft_x_fwd(const float* __restrict__ s1, const float* __restrict__ twx,
                          float* __restrict__ s2) {
  int bc = blockIdx.x;
  __shared__ float sl[NP * 2 * NKY];
  const float* sp = s1 + (size_t)bc * NP * 2 * NKY;
  for (int i = threadIdx.x; i < NP * 2 * NKY; i += 256) sl[i] = sp[i];
  __syncthreads();
  for (int o = threadIdx.x; o < NKX * NKY; o += 256) {
    int kx = o / NKY, ky = o % NKY;
    float ar = 0.f, ai = 0.f;
    for (int x = 0; x < NP; ++x) {
      float hr = sl[x * 2 * NKY + ky * 2], hi = sl[x * 2 * NKY + ky * 2 + 1];
      float wr = twx[(x * NKX + kx) * 2], wi = twx[(x * NKX + kx) * 2 + 1];
      ar += hr * wr - hi * wi;
      ai += hr * wi + hi * wr;
    }
    s2[((size_t)bc * NKX * NKY + o) * 2] = ar;
    s2[((size_t)bc * NKX * NKY + o) * 2 + 1] = ai;
  }
}

// ---- per-mode 64x64 complex channel mix: out[b,o] = sum_i in[b,i]*w[i,o] ----
__global__ void mode_mix(const float* __restrict__ s2, const float* __restrict__ w1r,
                         const float* __restrict__ w1i, const float* __restrict__ w2r,
                         const float* __restrict__ w2i, float* __restrict__ s3) {
  int kx = blockIdx.x / NKY, ky = blockIdx.x % NKY;
  __shared__ float wr[WID * WID], wi[WID * WID];
  __shared__ float inr[BB * WID], ini[BB * WID];
  int mx = (kx < 20) ? kx : kx - 20;
  const float* Wr = (kx < 20) ? w1r : w2r;
  const float* Wi = (kx < 20) ? w1i : w2i;
  for (int n = threadIdx.x; n < WID * WID; n += 256) {   // n = i*64+o ; stride 400 in weights
    size_t widx = ((size_t)n * NKY + mx) * NKY + ky;
    wr[n] = Wr[widx]; wi[n] = Wi[widx];
  }
  for (int n = threadIdx.x; n < BB * WID; n += 256) {
    int b = n >> 6, i = n & 63;
    size_t sidx = (((size_t)(b * WID + i) * NKX + kx) * NKY + ky) * 2;
    inr[n] = s2[sidx]; ini[n] = s2[sidx + 1];
  }
  __syncthreads();
  for (int n = threadIdx.x; n < BB * WID; n += 256) {
    int b = n >> 6, o = n & 63;
    float ar = 0.f, ai = 0.f;
    for (int i = 0; i < WID; ++i) {
      float hr = inr[b * WID + i], hi = ini[b * WID + i];
      float cr = wr[i * WID + o], ci = wi[i * WID + o];
      ar += hr * cr - hi * ci;
      ai += hr * ci + hi * cr;
    }
    size_t oidx = (((size_t)(b * WID + o) * NKX + kx) * NKY + ky) * 2;
    s3[oidx] = ar; s3[oidx + 1] = ai;
  }
}

// ---- inverse x-DFT: (40 kx, 20 ky) -> (268 x, 20 ky); packed [b][x][c][re0..19,im0..19] ----
__global__ void dft_x_inv(const float* __restrict__ s3, const float* __restrict__ tix,
                          float* __restrict__ s4) {
  int bc = blockIdx.x; int b = bc >> 6, c = bc & 63;
  __shared__ float sl[NKX * NKY * 2];
  const float* sp = s3 + (size_t)bc * NKX * NKY * 2;
  for (int i = threadIdx.x; i < NKX * NKY * 2; i += 256) sl[i] = sp[i];
  __syncthreads();
  for (int o = threadIdx.x; o < NP * NKY; o += 256) {
    int x = o / NKY, ky = o % NKY;
    float ar = 0.f, ai = 0.f;
    for (int kx = 0; kx < NKX; ++kx) {
      float hr = sl[(kx * NKY + ky) * 2], hi = sl[(kx * NKY + ky) * 2 + 1];
      float wr = tix[(x * NKX + kx) * 2], wi = tix[(x * NKX + kx) * 2 + 1];
      ar += hr * wr - hi * wi;
      ai += hr * wi + hi * wr;
    }
    size_t base = (((size_t)b * NP + x) * WID + c) * (2 * NKY);
    s4[base + ky] = ar;
    s4[base + NKY + ky] = ai;
  }
}

// ---- fused: hout = GELU( cw@h  +  S4@Tinv_t  + cb ) ; two WMMA GEMMs, one accumulator ----
// block = (b, x, t-half of 144 cols); 128 threads = 4 waves; M=64 (o), N=144 tile, K=64 / K=40
__global__ void spectral_fuse(const float* __restrict__ h, const float* __restrict__ s4,
                              const float* __restrict__ cw, const float* __restrict__ cb,
                              const float* __restrict__ tit, float* __restrict__ hout) {
  int bid = blockIdx.x;
  int half = bid & 1; int x = (bid >> 1) % NP; int b = bid / (NP * 2);
  int tbase = half * 144;
  __shared__ float sB[WID * 144];    // h slab [i][t], ld 144, zero-padded
  __shared__ float sA1[WID * 2 * NKY]; // packed spectra [o? no: rows are channels][40]
  __shared__ float sA2[WID * WID];   // cw layer slab
  for (int i = threadIdx.x; i < WID * 144; i += 128) {
    int c = i / 144, t = tbase + (i % 144);
    sB[i] = (t < NP) ? h[(((size_t)(b * WID + c)) * NP + x) * NP + t] : 0.f;
  }
  const float* a1src = s4 + (((size_t)b * NP + x) * WID) * (2 * NKY);
  for (int i = threadIdx.x; i < WID * 2 * NKY; i += 128) sA1[i] = a1src[i];
  for (int i = threadIdx.x; i < WID * WID; i += 128) sA2[i] = cw[i];
  __syncthreads();
  int wave = threadIdx.x >> 5, lane = threadIdx.x & 31;
  int mt = wave;                                   // 4 waves -> 4 M-tiles of 16
  for (int nt = 0; nt < 9; ++nt) {
    v8f acc = {};
#pragma unroll
    for (int k = 0; k < WID; k += 4)               // x2 = cw @ h
      acc = wmma4(frag_a(sA2, mt * 16, k, WID), frag_b(sB, k, nt * 16, 144), acc);
#pragma unroll
    for (int k = 0; k < 2 * NKY; k += 4)           // x1 = S4 @ Tinv_t (fused inverse t-DFT)
      acc = wmma4(frag_a(sA1, mt * 16, k, 2 * NKY), frag_b(tit, k, tbase + nt * 16, TIT_LD), acc);
    int col = tbase + nt * 16 + (lane & 15);
    if (col < NP) {
#pragma unroll
      for (int v = 0; v < 8; ++v) {
        int row = mt * 16 + v + ((lane >> 4) << 3);
        float val = acc[v] + cb[row];
        hout[(((size_t)(b * WID + row)) * NP + x) * NP + col] = gelu_exact(val);
      }
    }
  }
}

// ---- fused MLP head: crop + (64->128->32->1) per pixel; WMMA on transposed GEMMs ----
__global__ void mlp_head(const float* __restrict__ h, const float* __restrict__ w1,
                         const float* __restrict__ b1, const float* __restrict__ w2,
                         const float* __restrict__ b2, const float* __restrict__ w3,
                         const float* __restrict__ b3, float* __restrict__ out) {
  int bid = blockIdx.x;                 // b*1024 + x*4 + tt
  int tt = bid & 3, x = (bid >> 2) & 255, b = bid >> 10;
  int t0 = tt * 64;
  __shared__ float s0[WID * 64];        // in^T [c][p]
  __shared__ float sh1[128 * 64];       // h1   [n][p]
  __shared__ float sh2[32 * 64];        // h2   [n][p]
  for (int i = threadIdx.x; i < WID * 64; i += 256) {
    int c = i >> 6, p = i & 63;
    s0[i] = h[(((size_t)(b * WID + c)) * NP + x) * NP + t0 + p];
  }
  __syncthreads();
  int wave = threadIdx.x >> 5, lane = threadIdx.x & 31;
  for (int nt = 0; nt < 4; ++nt) {      // layer1: M=128 (w1 rows), N=64 pixels, K=64
    v8f acc = {};
#pragma unroll
    for (int k = 0; k < 64; k += 4)
      acc = wmma4(frag_a(w1, wave * 16, k, 64), frag_b(s0, k, nt * 16, 64), acc);
    int col = nt * 16 + (lane & 15);
#pragma unroll
    for (int v = 0; v < 8; ++v) {
      int row = wave * 16 + v + ((lane >> 4) << 3);
      sh1[row * 64 + col] = acc[v] + b1[row];
    }
  }
  __syncthreads();
  {                                     // layer2: M=32, N=64, K=128
    int mt2 = wave >> 2, nt2 = wave & 3;
    v8f acc = {};
#pragma unroll
    for (int k = 0; k < 128; k += 4)
      acc = wmma4(frag_a(w2, mt2 * 16, k, 128), frag_b(sh1, k, nt2 * 16, 64), acc);
    int col = nt2 * 16 + (lane & 15);
#pragma unroll
    for (int v = 0; v < 8; ++v) {
      int row = mt2 * 16 + v + ((lane >> 4) << 3);
      sh2[row * 64 + col] = acc[v] + b2[row];
    }
  }
  __syncthreads();
  if (threadIdx.x < 64) {               // layer3: 32 -> 1
    int p = threadIdx.x;
    float acc = b3[0];
#pragma unroll
    for (int n = 0; n < 32; ++n) acc += w3[n] * sh2[n * 64 + p];
    out[((size_t)b * 256 + x) * 256 + t0 + p] = acc;
  }
}

extern "C" void kernel_launch(void* const* d_in, const int* in_sizes, int n_in,
                              void* d_out, int out_size, void* d_ws, size_t ws_size,
                              hipStream_t stream) {
  (void)in_sizes; (void)n_in; (void)out_size; (void)ws_size;
  const float* xin    = (const float*)d_in[0];
  const float* wi_w   = (const float*)d_in[1];
  const float* wi_b   = (const float*)d_in[2];
  const float* w0_w   = (const float*)d_in[3];
  const float* w0_b   = (const float*)d_in[4];
  const float* sw1_re = (const float*)d_in[5];
  const float* sw1_im = (const float*)d_in[6];
  const float* sw2_re = (const float*)d_in[7];
  const float* sw2_im = (const float*)d_in[8];
  const float* cw     = (const float*)d_in[9];
  const float* cb     = (const float*)d_in[10];
  const float* fc1_w  = (const float*)d_in[11];
  const float* fc1_b  = (const float*)d_in[12];
  const float* fc2_w  = (const float*)d_in[13];
  const float* fc2_b  = (const float*)d_in[14];
  const float* fc3_w  = (const float*)d_in[15];
  const float* fc3_b  = (const float*)d_in[16];
  float* outp = (float*)d_out;

  // workspace carve-up (floats)
  const size_t NPH = (size_t)BB * WID * NP * NP;          // 36,773,888
  float* ws = (float*)d_ws;
  size_t off = 0;
  float* hA  = ws + off; off += NPH;
  float* hB  = ws + off; off += NPH;
  float* s1  = ws + off; off += (size_t)BB * WID * NP * 2 * NKY;
  float* s2  = ws + off; off += (size_t)BB * WID * NKX * NKY * 2;
  float* s3  = ws + off; off += (size_t)BB * WID * NKX * NKY * 2;
  float* s4  = ws + off; off += (size_t)BB * NP * WID * 2 * NKY;
  float* tft = ws + off; off += (size_t)NP * NKY * 2;
  float* tfx = ws + off; off += (size_t)NP * NKX * 2;
  float* tix = ws + off; off += (size_t)NP * NKX * 2;
  float* tit = ws + off; off += (size_t)NKX * TIT_LD;

  twiddle_init<<<(NKX * TIT_LD + 255) / 256, 256, 0, stream>>>(tft, tfx, tix, tit);
  lift_kernel<<<BB * NP, 128, 0, stream>>>(xin, wi_w, wi_b, w0_w, w0_b, hA);

  float* hcur = hA; float* hnxt = hB;
  for (int l = 0; l < 4; ++l) {
    size_t woff = (size_t)l * WID * WID * NKY * NKY;
    dft_t_fwd<<<BB * WID * NP, 64, 0, stream>>>(hcur, tft, s1);
    dft_x_fwd<<<BB * WID, 256, 0, stream>>>(s1, tfx, s2);
    mode_mix<<<NKX * NKY, 256, 0, stream>>>(s2, sw1_re + woff, sw1_im + woff,
                                            sw2_re + woff, sw2_im + woff, s3);
    dft_x_inv<<<BB * WID, 256, 0, stream>>>(s3, tix, s4);
    spectral_fuse<<<BB * NP * 2, 128, 0, stream>>>(hcur, s4, cw + (size_t)l * WID * WID,
                                                   cb + (size_t)l * WID, tit, hnxt);
    float* tmp = hcur; hcur = hnxt; hnxt = tmp;
  }
  mlp_head<<<BB * 256 * 4, 256, 0, stream>>>(hcur, fc1_w, fc1_b, fc2_w, fc2_b,
                                             fc3_w, fc3_b, outp);
}